// SelfAttention_20280835571985
// MI455X (gfx1250) — compile-verified
//
#include <hip/hip_runtime.h>

// ---- shapes ----
#define DIMK   2048
#define NH     32
#define NKV    8
#define HD     64
#define SEQ    2048
#define BATCH  2

typedef __attribute__((ext_vector_type(2))) float v2f;
typedef __attribute__((ext_vector_type(8))) float v8f;

// D = A(16x4) * B(4x16) + C(16x16), all f32, wave32.
__device__ __forceinline__ v8f wmma4(v2f a, v2f b, v8f c) {
    return __builtin_amdgcn_wmma_f32_16x16x4_f32(
        /*neg_a=*/false, a, /*neg_b=*/false, b,
        /*c_mod=*/(short)0, c, /*reuse_a=*/false, /*reuse_b=*/false);
}

// One wave computes a 16(M) x 64(N) f32 GEMM tile: acc[j] covers columns j*16..j*16+15.
// A: [M x K] row-major, lda;  B: [K x N] row-major, ldb.
__device__ __forceinline__ void gemm16x64(const float* __restrict__ A, int lda,
                                          const float* __restrict__ B, int ldb,
                                          int m0, int col0, int K, v8f acc[4]) {
    const int lane = threadIdx.x & 31;
    const int ln   = lane & 15;
    const int g2   = (lane >> 4) * 2;
    const float* pa = A + (size_t)(m0 + ln) * lda + g2;     // float2 @ k
    const float* pb = B + (size_t)g2 * ldb + col0 + ln;     // rows g2, g2+1 @ k
    acc[0] = {}; acc[1] = {}; acc[2] = {}; acc[3] = {};
    for (int k = 0; k < K; k += 4) {
        v2f a = *(const v2f*)(pa + k);
        const float* pbk = pb + (size_t)k * ldb;
#pragma unroll
        for (int j = 0; j < 4; ++j) {
            v2f b = { pbk[j * 16], pbk[j * 16 + ldb] };
            acc[j] = wmma4(a, b, acc[j]);
        }
    }
}

// ---------------- Kernel 1: fused QKV projection + RoPE ----------------
// grid (256 token tiles, 12), block 128 (4 waves). nt = 0..31 -> Q head, 32..39 -> K head, 40..47 -> V head.
__global__ void qkv_rope_kernel(const float* __restrict__ x,
                                const float* __restrict__ wq,
                                const float* __restrict__ wk,
                                const float* __restrict__ wv,
                                const float* __restrict__ fcos,
                                const float* __restrict__ fsin,
                                float* __restrict__ qws,   // [B,NH,SEQ,HD]
                                float* __restrict__ ktws,  // [B,NKV,HD,SEQ]
                                float* __restrict__ vws) { // [B,NKV,SEQ,HD]
    const int lane = threadIdx.x & 31;
    const int ln   = lane & 15;
    const int grp  = lane >> 4;
    const int m0   = blockIdx.x * 16;                 // flat token tile
    const int nt   = blockIdx.y * 4 + (threadIdx.x >> 5);

    const float* w; int ldw, h, mode;
    if (nt < 32)      { w = wq; ldw = NH  * HD; h = nt;      mode = 0; }
    else if (nt < 40) { w = wk; ldw = NKV * HD; h = nt - 32; mode = 1; }
    else              { w = wv; ldw = NKV * HD; h = nt - 40; mode = 2; }

    v8f acc[4];
    gemm16x64(x, DIMK, w, ldw, m0, h * HD, DIMK, acc);

    const int b  = m0 >> 11;          // tiles never cross batch (2048 % 16 == 0)
    const int s0 = m0 & (SEQ - 1);

    if (mode != 2) {
        // RoPE: even/odd dims are adjacent lanes -> shfl_xor(1)
#pragma unroll
        for (int j = 0; j < 4; ++j) {
            const int nhalf = j * 8 + (ln >> 1);
#pragma unroll
            for (int r = 0; r < 8; ++r) {
                const int pos = s0 + r + grp * 8;
                const float c = fcos[pos * (HD / 2) + nhalf];
                const float s = fsin[pos * (HD / 2) + nhalf];
                float vv = acc[j][r];
                float other = __shfl_xor(vv, 1, 32);
                acc[j][r] = (lane & 1) ? (other * s + vv * c)
                                       : (vv * c - other * s);
            }
        }
    }

    if (mode == 0) {
        float* base = qws + ((size_t)(b * NH + h) * SEQ + s0) * HD;
#pragma unroll
        for (int r = 0; r < 8; ++r) {
            const int m = r + grp * 8;
#pragma unroll
            for (int j = 0; j < 4; ++j)
                base[(size_t)m * HD + j * 16 + ln] = acc[j][r];
        }
    } else if (mode == 1) {
        float* base = ktws + (size_t)(b * NKV + h) * HD * SEQ;
#pragma unroll
        for (int r = 0; r < 8; ++r) {
            const int m = r + grp * 8;
#pragma unroll
            for (int j = 0; j < 4; ++j)
                base[(size_t)(j * 16 + ln) * SEQ + s0 + m] = acc[j][r];
        }
    } else {
        float* base = vws + ((size_t)(b * NKV + h) * SEQ + s0) * HD;
#pragma unroll
        for (int r = 0; r < 8; ++r) {
            const int m = r + grp * 8;
#pragma unroll
            for (int j = 0; j < 4; ++j)
                base[(size_t)m * HD + j * 16 + ln] = acc[j][r];
        }
    }
}

// ---------------- Kernel 2: causal flash attention ----------------
// grid (SEQ/16, NH, B), block 32 (one wave per 16-query tile of one head).
__global__ void attn_kernel(const float* __restrict__ qws,
                            const float* __restrict__ ktws,
                            const float* __restrict__ vws,
                            float* __restrict__ aws) {   // [B*SEQ, NH*HD]
    __shared__ float pl[16 * 18];
    const int lane = threadIdx.x & 31;
    const int ln   = lane & 15;
    const int grp  = lane >> 4;
    const int g2   = grp * 2;
    const int q0   = blockIdx.x * 16;
    const int h    = blockIdx.y;
    const int b    = blockIdx.z;
    const int hk   = h >> 2;                       // N_REP = 4

    const float* qb  = qws + ((size_t)(b * NH + h) * SEQ + q0) * HD;
    const float* ktb = ktws + (size_t)(b * NKV + hk) * HD * SEQ;
    const float* vb  = vws + (size_t)(b * NKV + hk) * SEQ * HD;

    // Preload Q tile in WMMA A-layout: a[v] = Q[ln][kk*4 + g2 + v]
    v2f aq[16];
#pragma unroll
    for (int kk = 0; kk < 16; ++kk)
        aq[kk] = *(const v2f*)(qb + ln * HD + kk * 4 + g2);

    v8f o[4]; o[0] = {}; o[1] = {}; o[2] = {}; o[3] = {};
    float mrow[8], lrow[8];
#pragma unroll
    for (int r = 0; r < 8; ++r) { mrow[r] = -1e30f; lrow[r] = 0.0f; }

    for (int n0 = 0; n0 <= q0; n0 += 16) {
        // ---- scores S = Q * K^T (16x16), K supplied from transposed buffer
        v8f s = {};
#pragma unroll
        for (int kk = 0; kk < 16; ++kk) {
            const float* kp = ktb + (size_t)(kk * 4 + g2) * SEQ + n0 + ln;
            v2f bk = { kp[0], kp[SEQ] };
            s = wmma4(aq[kk], bk, s);
        }

        const bool diag = (n0 == q0);
        float p[8];
#pragma unroll
        for (int r = 0; r < 8; ++r) {
            float sv = s[r] * 0.125f;                       // 1/sqrt(64)
            if (diag && (ln > r + grp * 8)) sv = -1e30f;    // causal mask
            // row max across the 16 lanes of this half-group
            float rm = sv;
            rm = fmaxf(rm, __shfl_xor(rm, 1, 32));
            rm = fmaxf(rm, __shfl_xor(rm, 2, 32));
            rm = fmaxf(rm, __shfl_xor(rm, 4, 32));
            rm = fmaxf(rm, __shfl_xor(rm, 8, 32));
            const float mn    = fmaxf(mrow[r], rm);
            const float alpha = expf(mrow[r] - mn);
            const float pv    = expf(sv - mn);
            float rs = pv;
            rs += __shfl_xor(rs, 1, 32);
            rs += __shfl_xor(rs, 2, 32);
            rs += __shfl_xor(rs, 4, 32);
            rs += __shfl_xor(rs, 8, 32);
            lrow[r] = lrow[r] * alpha + rs;
            mrow[r] = mn;
            p[r] = pv;
#pragma unroll
            for (int j = 0; j < 4; ++j) o[j][r] *= alpha;
        }

        // ---- re-layout P (C/D layout) -> A layout via LDS
        __syncthreads();
#pragma unroll
        for (int r = 0; r < 8; ++r)
            pl[(r + grp * 8) * 18 + ln] = p[r];
        __syncthreads();
        v2f ap[4];
#pragma unroll
        for (int kk = 0; kk < 4; ++kk)
            ap[kk] = *(const v2f*)(&pl[ln * 18 + kk * 4 + g2]);

        // ---- O += P * V  (16x16 * 16x64)
#pragma unroll
        for (int kk = 0; kk < 4; ++kk) {
            const float* vp = vb + (size_t)(n0 + kk * 4 + g2) * HD + ln;
#pragma unroll
            for (int j = 0; j < 4; ++j) {
                v2f bv = { vp[j * 16], vp[HD + j * 16] };
                o[j] = wmma4(ap[kk], bv, o[j]);
            }
        }
    }

    // normalize and store to [token][NH*HD]
    float* ob = aws + ((size_t)(b * SEQ) + q0) * (NH * HD) + h * HD;
#pragma unroll
    for (int r = 0; r < 8; ++r) {
        const float inv = 1.0f / lrow[r];
        const int m = r + grp * 8;
#pragma unroll
        for (int j = 0; j < 4; ++j)
            ob[(size_t)m * (NH * HD) + j * 16 + ln] = o[j][r] * inv;
    }
}

// ---------------- Kernel 3: output projection ----------------
// grid (256, 8), block 128 (4 waves): out = attn[4096x2048] @ wo[2048x2048]
__global__ void oproj_kernel(const float* __restrict__ aws,
                             const float* __restrict__ wo,
                             float* __restrict__ out) {
    const int lane = threadIdx.x & 31;
    const int ln   = lane & 15;
    const int grp  = lane >> 4;
    const int m0   = blockIdx.x * 16;
    const int nt   = blockIdx.y * 4 + (threadIdx.x >> 5);

    v8f acc[4];
    gemm16x64(aws, DIMK, wo, DIMK, m0, nt * 64, DIMK, acc);

#pragma unroll
    for (int r = 0; r < 8; ++r) {
        const int m = r + grp * 8;
#pragma unroll
        for (int j = 0; j < 4; ++j)
            out[(size_t)(m0 + m) * DIMK + nt * 64 + j * 16 + ln] = acc[j][r];
    }
}

extern "C" void kernel_launch(void* const* d_in, const int* in_sizes, int n_in,
                              void* d_out, int out_size, void* d_ws, size_t ws_size,
                              hipStream_t stream) {
    const float* x    = (const float*)d_in[0];
    // d_in[1] = start_pos (0, ignored), d_in[4] = mask (causal handled in-kernel)
    const float* fcos = (const float*)d_in[2];
    const float* fsin = (const float*)d_in[3];
    const float* wq   = (const float*)d_in[5];
    const float* wk   = (const float*)d_in[6];
    const float* wv   = (const float*)d_in[7];
    const float* wo   = (const float*)d_in[8];
    float* out = (float*)d_out;

    float* qws  = (float*)d_ws;                                    // B*NH*SEQ*HD
    float* ktws = qws  + (size_t)BATCH * NH  * SEQ * HD;           // B*NKV*HD*SEQ
    float* vws  = ktws + (size_t)BATCH * NKV * HD  * SEQ;          // B*NKV*SEQ*HD
    float* aws  = vws  + (size_t)BATCH * NKV * SEQ * HD;           // B*SEQ*NH*HD

    qkv_rope_kernel<<<dim3((BATCH * SEQ) / 16, 12, 1), 128, 0, stream>>>(
        x, wq, wk, wv, fcos, fsin, qws, ktws, vws);
    attn_kernel<<<dim3(SEQ / 16, NH, BATCH), 32, 0, stream>>>(
        qws, ktws, vws, aws);
    oproj_kernel<<<dim3((BATCH * SEQ) / 16, 8, 1), 128, 0, stream>>>(
        aws, wo, out);
}